// ImprovedRNN_60885456388737
// MI455X (gfx1250) — compile-verified
//
#include <hip/hip_runtime.h>

typedef __attribute__((ext_vector_type(2))) float v2f;
typedef __attribute__((ext_vector_type(8))) float v8f;

#define BB 256
#define TT 1024
#define II 64
#define HH 25
#define OO 64
#define G4 100            // 4*H
#define GP 112            // padded gate row stride
#define HP 32             // padded hidden row stride
#define ROWS (BB * TT)    // 262144

__device__ __forceinline__ float fast_sigmoid(float x) {
    return 1.0f / (1.0f + __expf(-x));
}
__device__ __forceinline__ float fast_tanh(float x) {
    // 1 - 2/(e^{2x}+1): saturates correctly to +-1 for large |x|
    float e = __expf(2.0f * x);
    return 1.0f - 2.0f / (e + 1.0f);
}

// ---------------------------------------------------------------------------
// Kernel 1: gates = x @ W_ih^T + (b_ih + b_hh)   -> gx [ROWS, GP]
// wave32 WMMA f32 16x16x4; each wave: 16 rows x 112 gate cols, K=64.
// Pad columns (100..111) get harmless garbage (never read) -> branchless.
// ---------------------------------------------------------------------------
__global__ __launch_bounds__(256) void lstm_gates_gemm(
    const float* __restrict__ x, const float* __restrict__ W_ih,
    const float* __restrict__ b_ih, const float* __restrict__ b_hh,
    float* __restrict__ gx)
{
    const int lane  = threadIdx.x & 31;
    const int wid   = threadIdx.x >> 5;
    const int mtile = blockIdx.x * 8 + wid;
    const int row0  = mtile * 16;
    const int mrow  = row0 + (lane & 15);
    const int koff  = (lane >> 4) * 2;
    const int ncol  = lane & 15;

    v8f acc[7] = {};

    #pragma unroll
    for (int ks = 0; ks < 16; ++ks) {
        const int k0 = ks * 4 + koff;
        v2f a;
        const float* ap = x + (size_t)mrow * II + k0;
        a.x = ap[0];
        a.y = ap[1];
        #pragma unroll
        for (int nt = 0; nt < 7; ++nt) {
            const int n  = nt * 16 + ncol;
            const int nc = (n < G4) ? n : 0;    // index clamp: no EXEC branches
            v2f b;
            b.x = W_ih[nc * II + k0];
            b.y = W_ih[nc * II + k0 + 1];
            acc[nt] = __builtin_amdgcn_wmma_f32_16x16x4_f32(
                false, a, false, b, (short)0, acc[nt], false, false);
        }
    }

    const int mbase = row0 + (lane >> 4) * 8;
    #pragma unroll
    for (int nt = 0; nt < 7; ++nt) {
        const int n  = nt * 16 + ncol;
        const int nc = (n < G4) ? n : 0;
        const float bias = b_ih[nc] + b_hh[nc];
        #pragma unroll
        for (int v = 0; v < 8; ++v) {
            gx[(size_t)(mbase + v) * GP + n] = acc[nt][v] + bias;
        }
    }
}

// ---------------------------------------------------------------------------
// Kernel 2: sequential LSTM scan. ONE WAVE (32 threads) per batch row.
// Single-wave workgroup => barrier ops are S_NOP (ISA 3.1): per-step sync is
// free. Thread u owns unit u and computes all 4 gate dots itself (4
// independent 25-FMA chains, no gate exchange). h broadcast via LDS
// (consecutive reads vectorize to ds_load_b128). Next step's gx prefetched.
// Writes padded hidden states hs [ROWS, HP] (cols 25..31 zero).
// ---------------------------------------------------------------------------
__global__ __launch_bounds__(32) void lstm_scan(
    const float* __restrict__ gx, const float* __restrict__ W_hh,
    float* __restrict__ hs)
{
    __shared__ float h_sh[HP];

    const int b  = blockIdx.x;
    const int u  = threadIdx.x;            // 0..31, units 25..31 are pad lanes
    const int uu = (u < HH) ? u : 0;       // clamp: pad lanes shadow unit 0

    // Per-thread gate weight rows (i, f, g, o) in registers: 100 VGPRs.
    float wi[HH], wf[HH], wg[HH], wo[HH];
    #pragma unroll
    for (int j = 0; j < HH; ++j) {
        wi[j] = W_hh[(uu     ) * HH + j];
        wf[j] = W_hh[(uu + 25) * HH + j];
        wg[j] = W_hh[(uu + 50) * HH + j];
        wo[j] = W_hh[(uu + 75) * HH + j];
    }

    h_sh[u] = 0.0f;
    float c = 0.0f;
    __syncthreads();

    const float* gxrow = gx + (size_t)b * TT * GP;
    float*       hsrow = hs + (size_t)b * TT * HP;

    float gi = gxrow[uu];
    float gf = gxrow[uu + 25];
    float gg = gxrow[uu + 50];
    float go = gxrow[uu + 75];

    for (int t = 0; t < TT; ++t) {
        // prefetch next timestep's gate inputs while computing this step
        float ni = 0.0f, nf = 0.0f, ng = 0.0f, no = 0.0f;
        if (t + 1 < TT) {
            const float* gnext = gxrow + (size_t)(t + 1) * GP;
            ni = gnext[uu];
            nf = gnext[uu + 25];
            ng = gnext[uu + 50];
            no = gnext[uu + 75];
        }

        float ai = gi, af = gf, ag = gg, ao = go;
        #pragma unroll
        for (int j = 0; j < HH; ++j) {     // 4 independent FMA chains
            const float hv = h_sh[j];      // LDS broadcast
            ai = fmaf(hv, wi[j], ai);
            af = fmaf(hv, wf[j], af);
            ag = fmaf(hv, wg[j], ag);
            ao = fmaf(hv, wo[j], ao);
        }

        const float iv = fast_sigmoid(ai);
        const float fv = fast_sigmoid(af);
        const float gv = fast_tanh(ag);
        const float ov = fast_sigmoid(ao);
        c = fv * c + iv * gv;
        const float hv = ov * fast_tanh(c);

        __syncthreads();                   // S_NOP for single-wave workgroup
        if (u < HH) {
            h_sh[u] = hv;
            hsrow[(size_t)t * HP + u] = hv;
        } else {
            hsrow[(size_t)t * HP + u] = 0.0f;   // keep K-padding zero
        }
        __syncthreads();

        gi = ni; gf = nf; gg = ng; go = no;
    }
}

// ---------------------------------------------------------------------------
// Kernel 3: out = hs @ W_fc^T + b_fc   ([ROWS,32-padded-25] x [25,64])
// wave32 WMMA f32 16x16x4; each wave: 16 rows x 64 out cols, K=32.
// A-operand is exactly zero for k>=25, so B uses clamped (finite) weights
// there: 0*finite = 0 -> branchless, no EXEC manipulation.
// ---------------------------------------------------------------------------
__global__ __launch_bounds__(256) void lstm_out_gemm(
    const float* __restrict__ hs, const float* __restrict__ W_fc,
    const float* __restrict__ b_fc, float* __restrict__ out)
{
    const int lane  = threadIdx.x & 31;
    const int wid   = threadIdx.x >> 5;
    const int mtile = blockIdx.x * 8 + wid;
    const int row0  = mtile * 16;
    const int mrow  = row0 + (lane & 15);
    const int koff  = (lane >> 4) * 2;
    const int ncol  = lane & 15;

    v8f acc[4] = {};

    #pragma unroll
    for (int ks = 0; ks < 8; ++ks) {
        const int k0  = ks * 4 + koff;
        const int kc0 = (k0     < HH) ? k0     : (HH - 1);  // index clamp
        const int kc1 = (k0 + 1 < HH) ? k0 + 1 : (HH - 1);
        v2f a;
        const float* ap = hs + (size_t)mrow * HP + k0;
        a.x = ap[0];              // hs cols 25..31 are zero
        a.y = ap[1];
        #pragma unroll
        for (int nt = 0; nt < 4; ++nt) {
            const int n = nt * 16 + ncol;
            v2f bf;
            bf.x = W_fc[n * HH + kc0];
            bf.y = W_fc[n * HH + kc1];
            acc[nt] = __builtin_amdgcn_wmma_f32_16x16x4_f32(
                false, a, false, bf, (short)0, acc[nt], false, false);
        }
    }

    const int mbase = row0 + (lane >> 4) * 8;
    #pragma unroll
    for (int nt = 0; nt < 4; ++nt) {
        const int n = nt * 16 + ncol;
        const float bias = b_fc[n];
        #pragma unroll
        for (int v = 0; v < 8; ++v) {
            out[(size_t)(mbase + v) * OO + n] = acc[nt][v] + bias;
        }
    }
}

// ---------------------------------------------------------------------------
extern "C" void kernel_launch(void* const* d_in, const int* in_sizes, int n_in,
                              void* d_out, int out_size, void* d_ws, size_t ws_size,
                              hipStream_t stream) {
    const float* x    = (const float*)d_in[0];
    const float* W_ih = (const float*)d_in[1];
    const float* W_hh = (const float*)d_in[2];
    const float* b_ih = (const float*)d_in[3];
    const float* b_hh = (const float*)d_in[4];
    const float* W_fc = (const float*)d_in[5];
    const float* b_fc = (const float*)d_in[6];
    float* out = (float*)d_out;

    float* gx = (float*)d_ws;                        // [ROWS, GP]
    float* hs = gx + (size_t)ROWS * GP;              // [ROWS, HP]

    // Kernel 1: 262144 rows / (16 rows * 8 waves) = 2048 blocks
    lstm_gates_gemm<<<ROWS / 128, 256, 0, stream>>>(x, W_ih, b_ih, b_hh, gx);

    // Kernel 2: one single-wave block per batch row
    lstm_scan<<<BB, 32, 0, stream>>>(gx, W_hh, hs);

    // Kernel 3: 2048 blocks
    lstm_out_gemm<<<ROWS / 128, 256, 0, stream>>>(hs, W_fc, b_fc, out);
}